// GraphConv_50036368998987
// MI455X (gfx1250) — compile-verified
//
#include <hip/hip_runtime.h>
#include <stdint.h>

#define D_FEAT 32
#define TILE   256   // edges per pipeline stage
#define NBUF   2     // double-buffered stages per block
#define BLOCK  256   // 8 waves (wave32)

__global__ __launch_bounds__(BLOCK) void zero_kernel(float* __restrict__ out, int n4) {
    int i = blockIdx.x * BLOCK + threadIdx.x;
    if (i < n4) {
        ((float4*)out)[i] = make_float4(0.f, 0.f, 0.f, 0.f);
    }
}

__global__ __launch_bounds__(BLOCK) void scatter_kernel(
    const float*     __restrict__ input,   // [N_NODES, 32]
    const long long* __restrict__ sidx,    // [n_edges]
    const long long* __restrict__ tidx,    // [n_edges]
    const float*     __restrict__ enorm,   // [n_edges]
    const float*     __restrict__ esgn,    // [n_edges]
    float*           __restrict__ out,     // [N_NODES, 32]
    int n_edges)
{
    __shared__ long long sm_s[NBUF][TILE];
    __shared__ long long sm_t[NBUF][TILE];
    __shared__ float     sm_n[NBUF][TILE];
    __shared__ float     sm_g[NBUF][TILE];

    const int tid = threadIdx.x;
    const long long e0 = (long long)blockIdx.x * (NBUF * TILE);

    // ---- Issue async-LDS staging for BOTH tiles up front (8 async ops/wave) ----
    // Edge stream is touched exactly once -> non-temporal so it doesn't evict the
    // L2-resident feature table / output accumulator.
#pragma unroll
    for (int b = 0; b < NBUF; ++b) {
        long long e  = e0 + (long long)b * TILE + tid;
        long long ec = (e < n_edges) ? e : (long long)(n_edges - 1);  // clamp: safe addr

        uint32_t lds_s = (uint32_t)(uintptr_t)&sm_s[b][tid];
        uint32_t lds_t = (uint32_t)(uintptr_t)&sm_t[b][tid];
        uint32_t lds_n = (uint32_t)(uintptr_t)&sm_n[b][tid];
        uint32_t lds_g = (uint32_t)(uintptr_t)&sm_g[b][tid];
        uint32_t off8  = (uint32_t)(ec << 3);   // byte offset into int64 arrays
        uint32_t off4  = (uint32_t)(ec << 2);   // byte offset into f32 arrays

        asm volatile(
            "global_load_async_to_lds_b64 %0, %4, %6 th:TH_LOAD_NT\n\t"
            "global_load_async_to_lds_b64 %1, %4, %7 th:TH_LOAD_NT\n\t"
            "global_load_async_to_lds_b32 %2, %5, %8 th:TH_LOAD_NT\n\t"
            "global_load_async_to_lds_b32 %3, %5, %9 th:TH_LOAD_NT"
            :
            : "v"(lds_s), "v"(lds_t), "v"(lds_n), "v"(lds_g),
              "v"(off8), "v"(off4),
              "s"(sidx), "s"(tidx), "s"(enorm), "s"(esgn)
            : "memory");
    }

    const int c    = tid & 7;   // feature chunk 0..7 (4 floats each)
    const int egrp = tid >> 3;  // edge slot base 0..31

    // ---- Stage 0: wait only for the first 4 async ops (in-order completion),
    //      tile 1's copies remain in flight while we process tile 0. ----
    asm volatile("s_wait_asynccnt 0x4" ::: "memory");
    __syncthreads();

#pragma unroll
    for (int b = 0; b < NBUF; ++b) {
        if (b == 1) {
            // ---- Stage 1: drain remaining async ops, publish to all waves ----
            asm volatile("s_wait_asynccnt 0x0" ::: "memory");
            __syncthreads();
        }

        const long long base = e0 + (long long)b * TILE;
#pragma unroll
        for (int k = 0; k < 8; ++k) {
            const int el = egrp + k * 32;          // uniform across the 8-lane group
            const long long eg = base + el;
            if (eg < n_edges) {
                const long long s = sm_s[b][el];
                const long long t = sm_t[b][el];
                const float     w = sm_n[b][el] * sm_g[b][el];

                const float4 v = *(const float4*)(input + s * D_FEAT + (c << 2));
                float* o = out + t * D_FEAT + (c << 2);
                atomicAdd(o + 0, v.x * w);
                atomicAdd(o + 1, v.y * w);
                atomicAdd(o + 2, v.z * w);
                atomicAdd(o + 3, v.w * w);
            }
        }
    }
}

extern "C" void kernel_launch(void* const* d_in, const int* in_sizes, int n_in,
                              void* d_out, int out_size, void* d_ws, size_t ws_size,
                              hipStream_t stream) {
    const float*     input = (const float*)d_in[0];
    const long long* eidx  = (const long long*)d_in[1];
    const float*     enorm = (const float*)d_in[2];
    const float*     esgn  = (const float*)d_in[3];
    float*           out   = (float*)d_out;

    const int n_edges = in_sizes[1] / 2;
    const long long* sidx = eidx;
    const long long* tidx = eidx + n_edges;

    // 1) Zero the accumulator (harness poisons d_out; re-zero every call).
    const int n4 = out_size / 4;
    zero_kernel<<<(n4 + BLOCK - 1) / BLOCK, BLOCK, 0, stream>>>(out, n4);

    // 2) Gather-scale-scatter with double-buffered async-LDS edge staging.
    const int edges_per_blk = NBUF * TILE;
    const int nblk = (n_edges + edges_per_blk - 1) / edges_per_blk;
    scatter_kernel<<<nblk, BLOCK, 0, stream>>>(input, sidx, tidx, enorm, esgn, out, n_edges);
}